// KDTreeDistanceLoss_58377195487675
// MI455X (gfx1250) — compile-verified
//
#include <hip/hip_runtime.h>

typedef float v2f __attribute__((ext_vector_type(2)));
typedef float v8f __attribute__((ext_vector_type(8)));

namespace {
constexpr int   kN        = 16384;            // src points
constexpr int   kM        = 16384;            // tgt points
constexpr int   kTile     = 16;
constexpr int   kSrcTiles = kN / kTile;       // 1024
constexpr int   kTgtTiles = kM / kTile;       // 1024
constexpr int   kWaves    = 8;                // waves per block (wave32)
constexpr int   kIters    = kTgtTiles / kWaves; // 128, uniform trip count
constexpr float kMaxDist  = 1.0f;
// d_ws layout: [0, 4KB) = 1024 partial sums; [4KB, 260KB) = packed B fragments
constexpr size_t kBFragByteOff = 4096;
}

// Plain v_min_num_f32 without the canonicalize ops fminf drags in
// (all inputs are finite: products/sums of normal(0,1) coords).
__device__ __forceinline__ float min_f32(float a, float b) {
  float r;
  asm("v_min_num_f32 %0, %1, %2" : "=v"(r) : "v"(a), "v"(b));
  return r;
}

// ds_swizzle XOR-lane exchange (group-of-32 mode: and=0x1f, xor=OFF>>10).
template <int OFF>
__device__ __forceinline__ float swz(float v) {
  return __int_as_float(__builtin_amdgcn_ds_swizzle(__float_as_int(v), OFF));
}

// Pre-pack the WMMA B fragment for every tgt tile, in exact lane layout:
// tile jt occupies 64 floats; lane n (0-15) -> (tx,ty) = K0,K1;
// lane 16+n -> (tz, ||t||^2) = K2,K3.  Inner loop then needs ONE b64 load.
__global__ __launch_bounds__(256)
void kd_prep_kernel(const float* __restrict__ tgt, float* __restrict__ bfrag) {
  const int j  = blockIdx.x * 256 + threadIdx.x;   // tgt point id
  const float tx = tgt[3 * j + 0];
  const float ty = tgt[3 * j + 1];
  const float tz = tgt[3 * j + 2];
  const float t2 = tx * tx + ty * ty + tz * tz;
  const int jt = j >> 4;
  const int n  = j & 15;
  float* base = bfrag + (size_t)jt * 64;
  v2f lo = {tx, ty};
  v2f hi = {tz, t2};
  *(v2f*)(base + 2 * n)      = lo;   // lanes 0-15 slot
  *(v2f*)(base + 32 + 2 * n) = hi;   // lanes 16-31 slot
}

// One block per 16-row src tile; 8 waves stride over the 1024 tgt tiles.
// Per iteration: 1x global_load_b64 (B frag) + 1x v_wmma_f32_16x16x4_f32
// (D = -2 s.t + ||t||^2, C=0) + 8x v_min_num_f32 running min.
__global__ __launch_bounds__(kWaves * 32)
void kd_nn_min_kernel(const float* __restrict__ src,
                      const v2f* __restrict__ bfrag,
                      float* __restrict__ partial) {
  __shared__ float smin[kWaves][kTile];

  const int lane = threadIdx.x & 31;
  const int wave = threadIdx.x >> 5;
  const int half = lane >> 4;
  const int l16  = lane & 15;
  const int i0   = blockIdx.x * kTile;

  // A fragment (16x4 f32): row m = (-2sx, -2sy, -2sz, 1)
  const float* sp = src + 3 * (i0 + l16);
  const float sx = sp[0], sy = sp[1], sz = sp[2];
  v2f a;
  a.x = half ? (-2.0f * sz) : (-2.0f * sx);   // VGPR0 = K0 | K2
  a.y = half ? 1.0f         : (-2.0f * sy);   // VGPR1 = K1 | K3

  float vm[8];
#pragma unroll
  for (int r = 0; r < 8; ++r) vm[r] = 3.0e38f;

  // Uniform trip count -> scalar branch, EXEC stays all-1s around WMMA.
  const v2f* bp = bfrag + (size_t)wave * 32 + lane;
#pragma unroll 2
  for (int it = 0; it < kIters; ++it) {
    const v2f b = bp[it * (kWaves * 32)];     // this wave's tile stride = 2KB
    v8f c = {0.f, 0.f, 0.f, 0.f, 0.f, 0.f, 0.f, 0.f};
    v8f d = __builtin_amdgcn_wmma_f32_16x16x4_f32(
        /*neg_a=*/false, a, /*neg_b=*/false, b,
        /*c_mod=*/(short)0, c, /*reuse_a=*/false, /*reuse_b=*/false);
#pragma unroll
    for (int r = 0; r < 8; ++r) vm[r] = min_f32(vm[r], d[r]);
  }

  // Min over the 16 columns: XOR swizzles within each 16-lane half.
#pragma unroll
  for (int r = 0; r < 8; ++r) {
    vm[r] = min_f32(vm[r], swz<0x041F>(vm[r]));   // SWAPX1
    vm[r] = min_f32(vm[r], swz<0x081F>(vm[r]));   // SWAPX2
    vm[r] = min_f32(vm[r], swz<0x101F>(vm[r]));   // SWAPX4
    vm[r] = min_f32(vm[r], swz<0x201F>(vm[r]));   // SWAPX8
  }
  // lane 0 holds rows 0..7, lane 16 holds rows 8..15
  if (l16 == 0) {
#pragma unroll
    for (int r = 0; r < 8; ++r) smin[wave][half * 8 + r] = vm[r];
  }
  __syncthreads();

  if (threadIdx.x < kTile) {
    const int row = threadIdx.x;
    float m = smin[0][row];
#pragma unroll
    for (int w = 1; w < kWaves; ++w) m = fminf(m, smin[w][row]);
    const float* sq = src + 3 * (i0 + row);
    const float x = sq[0], y = sq[1], z = sq[2];
    float d2 = m + (x * x + y * y + z * z);   // add ||s||^2 after the min
    d2 = fmaxf(d2, 0.0f);                     // numerical floor
    float val = (d2 > kMaxDist) ? 0.0f : d2;  // clamp rule
#pragma unroll
    for (int mask = 1; mask < 16; mask <<= 1)
      val += __shfl_xor(val, mask, 32);       // sum this tile's 16 rows
    if (row == 0) partial[blockIdx.x] = val;
  }
}

// Deterministic single-block reduction of the 1024 per-tile partial sums.
__global__ __launch_bounds__(256)
void kd_reduce_kernel(const float* __restrict__ partial, float* __restrict__ out) {
  __shared__ float sdata[8];
  float s = 0.0f;
  for (int i = threadIdx.x; i < kSrcTiles; i += 256) s += partial[i];
#pragma unroll
  for (int mask = 1; mask < 32; mask <<= 1) s += __shfl_xor(s, mask, 32);
  const int wave = threadIdx.x >> 5;
  const int lane = threadIdx.x & 31;
  if (lane == 0) sdata[wave] = s;
  __syncthreads();
  if (threadIdx.x < 8) {
    float v = sdata[threadIdx.x];
#pragma unroll
    for (int mask = 1; mask < 8; mask <<= 1) v += __shfl_xor(v, mask, 32);
    if (threadIdx.x == 0) out[0] = v / (float)kN;
  }
}

extern "C" void kernel_launch(void* const* d_in, const int* in_sizes, int n_in,
                              void* d_out, int out_size, void* d_ws, size_t ws_size,
                              hipStream_t stream) {
  (void)in_sizes; (void)n_in; (void)out_size; (void)ws_size;
  const float* src = (const float*)d_in[0];   // (16384, 3) f32
  const float* tgt = (const float*)d_in[1];   // (16384, 3) f32
  float* out       = (float*)d_out;           // scalar f32
  float* partial   = (float*)d_ws;                                  // 4 KB
  float* bfrag     = (float*)((char*)d_ws + kBFragByteOff);         // 256 KB

  kd_prep_kernel<<<kM / 256, 256, 0, stream>>>(tgt, bfrag);
  kd_nn_min_kernel<<<kSrcTiles, kWaves * 32, 0, stream>>>(src, (const v2f*)bfrag,
                                                          partial);
  kd_reduce_kernel<<<1, 256, 0, stream>>>(partial, out);
}